// PDN_dense_cat_skip_6bro_pre_batch_act_max_test_38010460569937
// MI455X (gfx1250) — compile-verified
//
#include <hip/hip_runtime.h>
#include <hip/hip_bf16.h>

typedef __attribute__((ext_vector_type(2))) float v2f;
typedef __attribute__((ext_vector_type(8))) float v8f;

#define N_NODES   50000
#define N_EDGES   800000
#define F_IN      64
#define EDGE_F    16
#define C         128
#define N_GRAPHS  1024
#define N_CLASSES 2
#define BN_EPS    1e-5f

// Always-hardware fp32 atomic add (global_atomic_add_f32, no CAS loop).
__device__ __forceinline__ void atom_add_f32(float* p, float v) {
    unsafeAtomicAdd(p, v);
}

// ---------------------------------------------------------------------------
// Edge MLP: w[e] = sigmoid( relu(ea@W1 + b1) @ W2 + b2 )
// ---------------------------------------------------------------------------
__global__ __launch_bounds__(256)
void edge_weight_kernel(const float* __restrict__ ea,
                        const float* __restrict__ W1, const float* __restrict__ b1,
                        const float* __restrict__ W2, const float* __restrict__ b2,
                        float* __restrict__ w) {
    __shared__ float sW1[EDGE_F * EDGE_F];
    __shared__ float sb1[EDGE_F];
    __shared__ float sW2[EDGE_F];
    __shared__ float sb2;
    int tid = threadIdx.x;
    if (tid < EDGE_F * EDGE_F) sW1[tid] = W1[tid];
    if (tid < EDGE_F) { sb1[tid] = b1[tid]; sW2[tid] = W2[tid]; }
    if (tid == 0) sb2 = b2[0];
    __syncthreads();
    int e = blockIdx.x * blockDim.x + tid;
    if (e >= N_EDGES) return;
    float a[EDGE_F];
    const float4* ea4 = (const float4*)(ea + (size_t)e * EDGE_F);
#pragma unroll
    for (int i = 0; i < 4; ++i) {
        float4 v = ea4[i];
        a[4*i+0] = v.x; a[4*i+1] = v.y; a[4*i+2] = v.z; a[4*i+3] = v.w;
    }
    float s = sb2;
#pragma unroll
    for (int j = 0; j < EDGE_F; ++j) {
        float h = sb1[j];
#pragma unroll
        for (int i = 0; i < EDGE_F; ++i) h = fmaf(a[i], sW1[i * EDGE_F + j], h);
        h = fmaxf(h, 0.0f);
        s = fmaf(h, sW2[j], s);
    }
    w[e] = 1.0f / (1.0f + __expf(-s));
}

// ---------------------------------------------------------------------------
// Degree / norm
// ---------------------------------------------------------------------------
__global__ void fill_kernel(float* __restrict__ p, float v, long long n) {
    long long i = (long long)blockIdx.x * blockDim.x + threadIdx.x;
    if (i < n) p[i] = v;
}

__global__ void deg_scatter_kernel(const int* __restrict__ col,
                                   const float* __restrict__ w,
                                   float* __restrict__ deg) {
    int e = blockIdx.x * blockDim.x + threadIdx.x;
    if (e >= N_EDGES) return;
    atom_add_f32(&deg[col[e]], w[e]);
}

__global__ void dinv_kernel(float* __restrict__ deg) {
    int i = blockIdx.x * blockDim.x + threadIdx.x;
    if (i >= N_NODES) return;
    float d = deg[i];
    deg[i] = (d > 0.0f) ? rsqrtf(d) : 0.0f;
}

// ---------------------------------------------------------------------------
// WMMA fp32 GEMM:  out[N_NODES x 128] = concat(Ablocks) @ W
// Each block: 16 output rows; 8 waves each own a 16-col slice; K steps of 4.
// ---------------------------------------------------------------------------
struct APtrs { const float* p[8]; };

__global__ __launch_bounds__(256)
void gemm_wmma_kernel(APtrs A, int nblk, int kshift,
                      const float* __restrict__ W, float* __restrict__ out) {
    const int K = nblk << kshift;
    const int S = K + 4;                    // LDS row stride (bank-conflict pad)
    const int kmask = (1 << kshift) - 1;
    extern __shared__ float As[];           // 16 * S floats

    const int tid = threadIdx.x;
    const int m_base = blockIdx.x * 16;

    for (int idx = tid; idx < 16 * K; idx += 256) {
        int row = idx / K;
        int k   = idx - row * K;
        int b   = k >> kshift;
        int kk  = k & kmask;
        As[row * S + k] = A.p[b][(size_t)(m_base + row) * (kmask + 1) + kk];
    }
    __syncthreads();

    const int lane = tid & 31;
    const int wid  = tid >> 5;
    const int m    = lane & 15;
    const int hi   = lane >> 4;             // 0: lanes 0-15, 1: lanes 16-31
    const int koff = hi << 1;               // K offset {0,1} vs {2,3}
    const int n0   = wid * 16 + m;          // output column

    v8f c = {};
    for (int k0 = 0; k0 < K; k0 += 4) {
        v2f a, b;
        a.x = As[m * S + k0 + koff];
        a.y = As[m * S + k0 + koff + 1];
        b.x = W[(size_t)(k0 + koff)     * C + n0];
        b.y = W[(size_t)(k0 + koff + 1) * C + n0];
        c = __builtin_amdgcn_wmma_f32_16x16x4_f32(false, a, false, b,
                                                  (short)0, c, false, false);
    }

    const int mhi = hi * 8;
#pragma unroll
    for (int v = 0; v < 8; ++v)
        out[(size_t)(m_base + v + mhi) * C + n0] = c[v];
}

// ---------------------------------------------------------------------------
// acc = dinv^2 * xl (self-loop) + bias ; then scatter edges
// ---------------------------------------------------------------------------
__global__ void init_acc_kernel(const float* __restrict__ xl,
                                const float* __restrict__ dinv,
                                const float* __restrict__ bias,
                                float* __restrict__ acc) {
    long long i = (long long)blockIdx.x * blockDim.x + threadIdx.x;
    if (i >= (long long)N_NODES * C) return;
    int n = (int)(i >> 7);
    int col = (int)(i & 127);
    float d = dinv[n];
    acc[i] = d * d * xl[i] + bias[col];
}

__global__ __launch_bounds__(256)
void scatter_kernel(const int* __restrict__ row, const int* __restrict__ col,
                    const float* __restrict__ w, const float* __restrict__ dinv,
                    const float* __restrict__ xl, float* __restrict__ acc) {
    long long t = (long long)blockIdx.x * blockDim.x + threadIdx.x;
    if (t >= (long long)N_EDGES * 32) return;
    int e    = (int)(t >> 5);
    int lane = (int)(t & 31);
    int r = row[e], cc = col[e];
    float nrm = dinv[r] * w[e] * dinv[cc];
    float4 xv = *(const float4*)(xl + (size_t)r * C + lane * 4);
    float* dst = acc + (size_t)cc * C + lane * 4;
    atom_add_f32(dst + 0, nrm * xv.x);
    atom_add_f32(dst + 1, nrm * xv.y);
    atom_add_f32(dst + 2, nrm * xv.z);
    atom_add_f32(dst + 3, nrm * xv.w);
}

// ---------------------------------------------------------------------------
// BatchNorm (training-mode batch stats) + ReLU
// ---------------------------------------------------------------------------
__global__ void bn_zero_kernel(float* __restrict__ sums) {
    if (threadIdx.x < 2 * C) sums[threadIdx.x] = 0.0f;
}

__global__ __launch_bounds__(128)
void bn_partial_kernel(const float* __restrict__ x,
                       float* __restrict__ sum, float* __restrict__ sumsq) {
    int colid = threadIdx.x;                 // 128 threads: one column each
    int r0 = blockIdx.x * 256;
    int rend = min(r0 + 256, N_NODES);
    float s = 0.0f, s2 = 0.0f;
    for (int r = r0; r < rend; ++r) {
        float v = x[(size_t)r * C + colid];
        s += v; s2 = fmaf(v, v, s2);
    }
    atom_add_f32(&sum[colid], s);
    atom_add_f32(&sumsq[colid], s2);
}

__global__ void bn_final_kernel(const float* __restrict__ sum,
                                const float* __restrict__ sumsq,
                                float* __restrict__ mean, float* __restrict__ scale) {
    int c = threadIdx.x;
    if (c >= C) return;
    float m = sum[c] * (1.0f / N_NODES);
    float v = sumsq[c] * (1.0f / N_NODES) - m * m;
    mean[c]  = m;
    scale[c] = rsqrtf(v + BN_EPS);
}

__global__ void bn_relu_kernel(const float* __restrict__ x,
                               const float* __restrict__ mean,
                               const float* __restrict__ scale,
                               float* __restrict__ y) {
    long long i = (long long)blockIdx.x * blockDim.x + threadIdx.x;
    if (i >= (long long)N_NODES * C) return;
    int col = (int)(i & 127);
    y[i] = fmaxf((x[i] - mean[col]) * scale[col], 0.0f);
}

// ---------------------------------------------------------------------------
// Global max-pool per graph (post-ReLU values >= 0, so uint-bit atomicMax ok)
// ---------------------------------------------------------------------------
__global__ void pooled_init_kernel(float* __restrict__ pooled) {
    int i = blockIdx.x * blockDim.x + threadIdx.x;
    if (i < N_GRAPHS * C) pooled[i] = 0.0f;
}

__global__ void pool_max_kernel(const float* __restrict__ x,
                                const int* __restrict__ batch,
                                float* __restrict__ pooled) {
    long long i = (long long)blockIdx.x * blockDim.x + threadIdx.x;
    if (i >= (long long)N_NODES * C) return;
    int n = (int)(i >> 7);
    int col = (int)(i & 127);
    float v = fmaxf(x[i], 0.0f);
    atomicMax((unsigned int*)&pooled[(size_t)batch[n] * C + col], __float_as_uint(v));
}

__global__ void lin_kernel(const float* __restrict__ pooled,
                           const float* __restrict__ W, const float* __restrict__ b,
                           float* __restrict__ out) {
    int t = blockIdx.x * blockDim.x + threadIdx.x;
    if (t >= N_GRAPHS * N_CLASSES) return;
    int g = t / N_CLASSES, o = t - g * N_CLASSES;
    float s = b[o];
    for (int c = 0; c < C; ++c)
        s = fmaf(pooled[(size_t)g * C + c], W[c * N_CLASSES + o], s);
    out[t] = s;
}

// ---------------------------------------------------------------------------
// Host orchestration
// ---------------------------------------------------------------------------
struct PdnP { const float *W, *W1, *W2, *b, *b1, *b2; };

extern "C" void kernel_launch(void* const* d_in, const int* in_sizes, int n_in,
                              void* d_out, int out_size, void* d_ws, size_t ws_size,
                              hipStream_t stream) {
    const float* x_in      = (const float*)d_in[0];
    const int*   edge_idx  = (const int*)d_in[1];
    const int*   batch     = (const int*)d_in[2];
    // d_in[3] = dropout (identity in test mode)
    const float* edge_attr = (const float*)d_in[4];
    const int* rowp = edge_idx;
    const int* colp = edge_idx + N_EDGES;

    auto getp = [&](int base) -> PdnP {
        return { (const float*)d_in[base],     (const float*)d_in[base + 1],
                 (const float*)d_in[base + 2], (const float*)d_in[base + 3],
                 (const float*)d_in[base + 4], (const float*)d_in[base + 5] };
    };
    int pi = 5;
    PdnP conv1 = getp(pi); pi += 6;
    PdnP hidden[6][2];
    for (int k = 0; k < 6; ++k)
        for (int j = 0; j < 2; ++j) { hidden[k][j] = getp(pi); pi += 6; }
    const float* linW = (const float*)d_in[pi];
    const float* linb = (const float*)d_in[pi + 1]; pi += 2;
    PdnP pool[6];
    for (int k = 0; k < 6; ++k) { pool[k] = getp(pi); pi += 6; }

    // workspace carve (floats)
    float* ws = (float*)d_ws;
    const size_t NC = (size_t)N_NODES * C;
    float* skip[7];
    for (int i = 0; i < 7; ++i) skip[i] = ws + (size_t)i * NC;
    float* bufA    = ws + 7 * NC;
    float* bufB    = ws + 8 * NC;
    float* xbn     = ws + 9 * NC;
    float* xl      = ws + 10 * NC;
    float* wE      = ws + 11 * NC;
    float* dinv    = wE + N_EDGES;
    float* bnS     = dinv + N_NODES;      // 128 sums + 128 sumsq (contiguous)
    float* bnS2    = bnS + C;
    float* bnM     = bnS2 + C;
    float* bnScale = bnM + C;
    float* pooled  = bnScale + C;

    const long long NCt = (long long)N_NODES * C;
    const int gNC = (int)((NCt + 255) / 256);
    const int gE  = (N_EDGES + 255) / 256;
    const int gN  = (N_NODES + 255) / 256;

    auto run_conv = [&](const APtrs& A, int nblk, int kshift, const PdnP& P,
                        float* outbuf) {
        edge_weight_kernel<<<gE, 256, 0, stream>>>(edge_attr, P.W1, P.b1, P.W2, P.b2, wE);
        fill_kernel<<<gN, 256, 0, stream>>>(dinv, 1.0f, N_NODES);   // self-loop weight
        deg_scatter_kernel<<<gE, 256, 0, stream>>>(colp, wE, dinv);
        dinv_kernel<<<gN, 256, 0, stream>>>(dinv);
        const int K = nblk << kshift;
        size_t shmem = (size_t)16 * (K + 4) * sizeof(float);
        gemm_wmma_kernel<<<N_NODES / 16, 256, shmem, stream>>>(A, nblk, kshift, P.W, xl);
        init_acc_kernel<<<gNC, 256, 0, stream>>>(xl, dinv, P.b, outbuf);
        long long T = (long long)N_EDGES * 32;
        scatter_kernel<<<(int)((T + 255) / 256), 256, 0, stream>>>(rowp, colp, wE, dinv, xl, outbuf);
    };

    auto run_bn_relu = [&](const float* xp, float* yp) {
        bn_zero_kernel<<<1, 256, 0, stream>>>(bnS);
        bn_partial_kernel<<<(N_NODES + 255) / 256, 128, 0, stream>>>(xp, bnS, bnS2);
        bn_final_kernel<<<1, 128, 0, stream>>>(bnS, bnS2, bnM, bnScale);
        bn_relu_kernel<<<gNC, 256, 0, stream>>>(xp, bnM, bnScale, yp);
    };

    // conv1: 64 -> 128
    {
        APtrs A{}; A.p[0] = x_in;
        run_conv(A, 1, 6, conv1, skip[0]);
    }
    float* xcur = skip[0];

    for (int k = 0; k < 6; ++k) {
        for (int j = 0; j < 2; ++j) {
            run_bn_relu(xcur, xbn);
            float* outb = (j == 0) ? bufA : bufB;
            APtrs A{}; A.p[0] = xbn;
            run_conv(A, 1, 7, hidden[k][j], outb);
            xcur = outb;
        }
        APtrs A{};
        A.p[0] = xcur;
        for (int s = 0; s <= k; ++s) A.p[s + 1] = skip[s];
        run_conv(A, k + 2, 7, pool[k], skip[k + 1]);
        xcur = skip[k + 1];
    }

    pooled_init_kernel<<<(N_GRAPHS * C + 255) / 256, 256, 0, stream>>>(pooled);
    pool_max_kernel<<<gNC, 256, 0, stream>>>(xcur, batch, pooled);
    lin_kernel<<<(N_GRAPHS * N_CLASSES + 255) / 256, 256, 0, stream>>>(pooled, linW, linb, (float*)d_out);
}